// AtomicOrbitals_84954453115270
// MI455X (gfx1250) — compile-verified
//
#include <hip/hip_runtime.h>

#ifndef __has_builtin
#define __has_builtin(x) 0
#endif

#define NB      512
#define NELEC   100
#define NATOMS  20
#define SHPA    30
#define NORB    300
#define NBAS    600          // NORB * NCTR, NCTR = 2
#define TOTAL   (NB * NELEC * NORB)   // 15,360,000 outputs

// CDNA5 async global->LDS path (ASYNCcnt-tracked), guarded so the host pass
// and older toolchains fall back to plain LDS staging.
#if defined(__gfx1250__) && __has_builtin(__builtin_amdgcn_global_load_async_to_lds_b32)
#define USE_ASYNC_LDS 1
__device__ __forceinline__ void async_cp_dword(const void* g, void* l) {
    // Builtin takes generic int* params on this toolchain:
    // (global src, lds dst, imm offset, imm cpol)
    __builtin_amdgcn_global_load_async_to_lds_b32((int*)g, (int*)l, 0, 0);
}
#else
#define USE_ASYNC_LDS 0
#endif

__global__ __launch_bounds__(256) void AtomicOrbitals_kernel(
    const float* __restrict__ pos,         // (NB, NELEC*3)
    const float* __restrict__ atom_coords, // (NATOMS, 3)
    const float* __restrict__ bas_exp,     // (NBAS,)
    const float* __restrict__ bas_n,       // (NBAS,) float 0/1/2
    const float* __restrict__ norm_cst,    // (NBAS,)
    const float* __restrict__ bas_coeffs,  // (NBAS,)
    const int*   __restrict__ bas_kxyz,    // (NBAS, 3) ints in {0,1,2}
    float*       __restrict__ out)         // (NB, NELEC, NORB)
{
    __shared__ float s_alpha[NBAS];
    __shared__ float s_n[NBAS];
    __shared__ float s_norm[NBAS];
    __shared__ float s_coef[NBAS];
    __shared__ int   s_k[NBAS * 3];
    __shared__ float s_at[NATOMS * 3];

    const int tid = threadIdx.x;
    const int nthreads = blockDim.x;

#if USE_ASYNC_LDS
    for (int i = tid; i < NBAS; i += nthreads) {
        async_cp_dword(bas_exp    + i, s_alpha + i);
        async_cp_dword(bas_n      + i, s_n     + i);
        async_cp_dword(norm_cst   + i, s_norm  + i);
        async_cp_dword(bas_coeffs + i, s_coef  + i);
    }
    for (int i = tid; i < NBAS * 3; i += nthreads)
        async_cp_dword(bas_kxyz + i, s_k + i);
    for (int i = tid; i < NATOMS * 3; i += nthreads)
        async_cp_dword(atom_coords + i, s_at + i);
  #if __has_builtin(__builtin_amdgcn_s_wait_asynccnt)
    __builtin_amdgcn_s_wait_asynccnt(0);
  #else
    asm volatile("s_wait_asynccnt 0" ::: "memory");
  #endif
#else
    for (int i = tid; i < NBAS; i += nthreads) {
        s_alpha[i] = bas_exp[i];
        s_n[i]     = bas_n[i];
        s_norm[i]  = norm_cst[i];
        s_coef[i]  = bas_coeffs[i];
    }
    for (int i = tid; i < NBAS * 3; i += nthreads)
        s_k[i] = bas_kxyz[i];
    for (int i = tid; i < NATOMS * 3; i += nthreads)
        s_at[i] = atom_coords[i];
#endif
    __syncthreads();

    const int stride = gridDim.x * blockDim.x;
    for (int idx = blockIdx.x * blockDim.x + tid; idx < TOTAL; idx += stride) {
        const unsigned uidx = (unsigned)idx;
        const int orb = (int)(uidx % (unsigned)NORB);
        const int row = (int)(uidx / (unsigned)NORB);

        // Prefetch next iteration's electron position (global_prefetch_b8).
        const int nidx = idx + stride;
        if (nidx < TOTAL)
            __builtin_prefetch(pos + ((unsigned)nidx / (unsigned)NORB) * 3, 0, 0);

        const float px = pos[row * 3 + 0];
        const float py = pos[row * 3 + 1];
        const float pz = pos[row * 3 + 2];

        float acc = 0.0f;
        const int bas0 = orb * 2;
#pragma unroll
        for (int s = 0; s < 2; ++s) {
            const int bas  = bas0 + s;
            const int atom = (int)((unsigned)bas / (unsigned)SHPA);

            const float dx = px - s_at[atom * 3 + 0];
            const float dy = py - s_at[atom * 3 + 1];
            const float dz = pz - s_at[atom * 3 + 2];
            const float r2 = dx * dx + dy * dy + dz * dz;

            const float a  = s_alpha[bas];
            const float e  = __expf(-a * r2);          // v_exp_f32

            const float nf = s_n[bas];
            const float r  = __builtin_amdgcn_sqrtf(r2); // v_sqrt_f32
            const float R  = (nf < 0.5f) ? 1.0f : ((nf < 1.5f) ? r : r2);

            const int kx = s_k[bas * 3 + 0];
            const int ky = s_k[bas * 3 + 1];
            const int kz = s_k[bas * 3 + 2];
            const float Yx = (kx == 0) ? 1.0f : ((kx == 1) ? dx : dx * dx);
            const float Yy = (ky == 0) ? 1.0f : ((ky == 1) ? dy : dy * dy);
            const float Yz = (kz == 0) ? 1.0f : ((kz == 1) ? dz : dz * dz);

            const float c = s_norm[bas] * s_coef[bas];
            acc += c * R * e * (Yx * (Yy * Yz));
        }
        out[idx] = acc;
    }
}

extern "C" void kernel_launch(void* const* d_in, const int* in_sizes, int n_in,
                              void* d_out, int out_size, void* d_ws, size_t ws_size,
                              hipStream_t stream) {
    (void)in_sizes; (void)n_in; (void)d_ws; (void)ws_size; (void)out_size;
    const float* pos         = (const float*)d_in[0];
    const float* atom_coords = (const float*)d_in[1];
    const float* bas_exp     = (const float*)d_in[2];
    const float* bas_n       = (const float*)d_in[3];
    const float* norm_cst    = (const float*)d_in[4];
    const float* bas_coeffs  = (const float*)d_in[5];
    const int*   bas_kxyz    = (const int*)d_in[6];
    // d_in[7] (index_ctr) is the fixed repeat(arange(NORB), 2) pattern; folded
    // into the bas = 2*orb + s indexing above.
    float* out = (float*)d_out;

    dim3 grid(1024), block(256);
    AtomicOrbitals_kernel<<<grid, block, 0, stream>>>(
        pos, atom_coords, bas_exp, bas_n, norm_cst, bas_coeffs, bas_kxyz, out);
}